// GatedLSTM_7928509629225
// MI455X (gfx1250) — compile-verified
//
#include <hip/hip_runtime.h>
#include <math.h>

// ---------------------------------------------------------------------------
// fp32 GEMV primitives built on V_WMMA_F32_16X16X4_F32 (CDNA5 / gfx1250).
// A wave computes 16 outputs per row-tile: rows [n0,n0+16) of row-major
// W[rows x ldw] dotted with x[K].  A carries x in matrix row 0 (lanes 0/16
// hold K=k0..k0+3 per the 32-bit A 16x4 layout); B holds 16 weight rows,
// one b64 load per lane per WMMA = exactly the 256-B B operand.
// D row M=0 lands in acc[0] of lanes 0..15.
// To avoid one long serially-dependent WMMA accumulation chain (the latency
// bottleneck for the sequential LSTM phases), each helper interleaves
// multiple INDEPENDENT accumulator chains inside a single k-loop.
// ---------------------------------------------------------------------------
typedef float v2f __attribute__((ext_vector_type(2)));
typedef float v8f __attribute__((ext_vector_type(8)));

__device__ __forceinline__ float sigmoidf_(float x) {
    return 1.0f / (1.0f + expf(-x));
}

template <bool NT>
__device__ __forceinline__ v2f ldw2(const float* p) {
    if constexpr (NT) {
        return __builtin_nontemporal_load((const v2f*)p);
    } else {
        return *(const v2f*)p;
    }
}

__device__ __forceinline__ v2f mk_a(const float* __restrict__ x, int k, bool keep) {
    v2f a;
    float xa = x[k];
    float xb = x[k + 1];
    a.x = keep ? xa : 0.0f;
    a.y = keep ? xb : 0.0f;
    return a;
}

#define WMMA_F32X4(A, B, C)                                                    \
    __builtin_amdgcn_wmma_f32_16x16x4_f32(false, (A), false, (B), (short)0,    \
                                          (C), false, false)

// Single matrix, one 16-row tile, K split into two independent chains.
template <bool NT>
__device__ __forceinline__ float wmma_mv_split2(const float* __restrict__ W,
                                                int ldw, int n0,
                                                const float* __restrict__ x,
                                                int K) {
    const int lane = (int)(threadIdx.x & 31u);
    const int koff = (lane >> 4) << 1;      // 0 (lanes 0-15) / 2 (lanes 16-31)
    const bool keep = (lane & 15) == 0;     // lanes 0,16 carry A row M=0
    const int half = K >> 1;
    const float* w0 = W + (size_t)(n0 + (lane & 15)) * (size_t)ldw + koff;
    const float* w1 = w0 + half;
    v8f acc0 = {};
    v8f acc1 = {};
#pragma unroll 4
    for (int k0 = 0; k0 < half; k0 += 4) {
        v2f b0 = ldw2<NT>(w0 + k0);
        v2f b1 = ldw2<NT>(w1 + k0);
        v2f a0 = mk_a(x, k0 + koff, keep);
        v2f a1 = mk_a(x, half + k0 + koff, keep);
        acc0 = WMMA_F32X4(a0, b0, acc0);
        acc1 = WMMA_F32X4(a1, b1, acc1);
    }
    return acc0[0] + acc1[0];
}

// LSTM gate helper: two row-tiles (n0a, n0b) x two matrices (Wi·x + Wh·h)
// fused into one k-loop -> 4 independent WMMA chains per wave.
__device__ __forceinline__ void wave_gates2(const float* __restrict__ Wi,
                                            const float* __restrict__ Wh,
                                            int ldw, int n0a, int n0b,
                                            const float* __restrict__ x,
                                            const float* __restrict__ h, int K,
                                            float* ra, float* rb) {
    const int lane = (int)(threadIdx.x & 31u);
    const int koff = (lane >> 4) << 1;
    const bool keep = (lane & 15) == 0;
    const size_t rA = (size_t)(n0a + (lane & 15)) * (size_t)ldw + koff;
    const size_t rB = (size_t)(n0b + (lane & 15)) * (size_t)ldw + koff;
    const float* wia = Wi + rA;
    const float* wha = Wh + rA;
    const float* wib = Wi + rB;
    const float* whb = Wh + rB;
    v8f a0 = {};
    v8f a1 = {};
    v8f a2 = {};
    v8f a3 = {};
#pragma unroll 2
    for (int k0 = 0; k0 < K; k0 += 4) {
        v2f ax = mk_a(x, k0 + koff, keep);
        v2f ah = mk_a(h, k0 + koff, keep);
        v2f b0 = ldw2<false>(wia + k0);
        v2f b1 = ldw2<false>(wha + k0);
        v2f b2 = ldw2<false>(wib + k0);
        v2f b3 = ldw2<false>(whb + k0);
        a0 = WMMA_F32X4(ax, b0, a0);
        a1 = WMMA_F32X4(ah, b1, a1);
        a2 = WMMA_F32X4(ax, b2, a2);
        a3 = WMMA_F32X4(ah, b3, a3);
    }
    *ra = a0[0] + a1[0];
    *rb = a2[0] + a3[0];
}

// Output layout (fp32), 53072 elements total:
//   [0,      50000)  y
//   [50000,  51024)  h_word_h  (2 x 512)
//   [51024,  52048)  h_word_c  (2 x 512)
//   [52048,  52560)  h_char_h  (2 x 256)
//   [52560,  53072)  h_char_c  (2 x 256)
#define OFF_WH 50000
#define OFF_WC 51024
#define OFF_CH 52048
#define OFF_CC 52560

// ---------------------------------------------------------------------------
// Kernel 1: char LSTM, 12 steps x 2 layers (H=256).  Single workgroup,
// 32 waves; each wave computes two 16-row gate tiles per layer-step with
// 4 interleaved WMMA chains.
// ---------------------------------------------------------------------------
__global__ __launch_bounds__(1024) void char_lstm_kernel(
    const int* __restrict__ x_char,
    const float* __restrict__ h0, const float* __restrict__ c0,
    const float* __restrict__ emb,
    const float* __restrict__ Wih, const float* __restrict__ Whh,
    const float* __restrict__ bih, const float* __restrict__ bhh,
    float* __restrict__ out) {
    __shared__ float s_x[256];
    __shared__ float s_h[2][256];
    __shared__ float s_c[2][256];
    __shared__ float s_gates[1024];

    const int t = (int)threadIdx.x;
    const int wave = t >> 5;
    const int lane = t & 31;

    if (t < 256) {
        s_h[0][t] = h0[t];       s_h[1][t] = h0[256 + t];
        s_c[0][t] = c0[t];       s_c[1][t] = c0[256 + t];
    }
    __syncthreads();

    for (int step = 0; step < 12; ++step) {
        const int cidx = x_char[step];
        if (t < 256) s_x[t] = emb[cidx * 256 + t];
        __syncthreads();

        for (int l = 0; l < 2; ++l) {
            const float* inp = (l == 0) ? s_x : s_h[0];
            const float* Wi = Wih + (size_t)l * 1024 * 256;
            const float* Wh = Whh + (size_t)l * 1024 * 256;
            const int n0a = wave * 32;
            const int n0b = wave * 32 + 16;
            float ga, gb;
            wave_gates2(Wi, Wh, 256, n0a, n0b, inp, s_h[l], 256, &ga, &gb);
            if (lane < 16) {
                s_gates[n0a + lane] = ga;
                s_gates[n0b + lane] = gb;
            }
            __syncthreads();
            if (t < 256) {
                const int b0 = l * 1024;
                float gi = s_gates[t]       + bih[b0 + t]       + bhh[b0 + t];
                float gf = s_gates[256 + t] + bih[b0 + 256 + t] + bhh[b0 + 256 + t];
                float gg = s_gates[512 + t] + bih[b0 + 512 + t] + bhh[b0 + 512 + t];
                float go = s_gates[768 + t] + bih[b0 + 768 + t] + bhh[b0 + 768 + t];
                float i = sigmoidf_(gi);
                float f = sigmoidf_(gf);
                float g = tanhf(gg);
                float o = sigmoidf_(go);
                float c = f * s_c[l][t] + i * g;
                s_c[l][t] = c;
                s_h[l][t] = o * tanhf(c);
            }
            __syncthreads();
        }
    }

    if (t < 256) {
        out[OFF_CH + t]       = s_h[0][t];
        out[OFF_CH + 256 + t] = s_h[1][t];
        out[OFF_CC + t]       = s_c[0][t];
        out[OFF_CC + 256 + t] = s_c[1][t];
    }
}

// ---------------------------------------------------------------------------
// Kernel 2: x_c = char_to_embedding @ c_char[1]; scalar gate; mix;
// word LSTM 2 layers (H=512).  Single workgroup, 32 waves; each wave owns
// four 16-row gate tiles, processed as two fused tile-pairs (4 chains each).
// ---------------------------------------------------------------------------
__global__ __launch_bounds__(1024) void word_lstm_kernel(
    const int* __restrict__ x_word,
    const float* __restrict__ h0, const float* __restrict__ c0,
    const float* __restrict__ word_emb,
    const float* __restrict__ Wih, const float* __restrict__ Whh,
    const float* __restrict__ bih, const float* __restrict__ bhh,
    const float* __restrict__ c2e,   // [512 x 256] row-major
    const float* __restrict__ g_w, const float* __restrict__ g_b,
    float* __restrict__ out, float* __restrict__ y_ws) {
    __shared__ float s_x[512];
    __shared__ float s_xw[512];
    __shared__ float s_xc[512];
    __shared__ float s_h[2][512];
    __shared__ float s_c[2][512];
    __shared__ float s_gates[2048];
    __shared__ float s_red[512];

    const int t = (int)threadIdx.x;
    const int wave = t >> 5;
    const int lane = t & 31;
    const int widx = x_word[0];

    if (t < 512) {
        s_h[0][t] = h0[t];       s_h[1][t] = h0[512 + t];
        s_c[0][t] = c0[t];       s_c[1][t] = c0[512 + t];
        s_xw[t] = word_emb[(size_t)widx * 512 + t];
    }
    __syncthreads();

    // x_c = c2e @ c_char_last   (c of last char layer, written by kernel 1)
    {
        const float* cvec = out + OFF_CC + 256;
        const int n0 = wave * 16;
        float r = wmma_mv_split2<false>(c2e, 256, n0, cvec, 256);
        if (lane < 16) s_xc[n0 + lane] = r;
    }
    if (t < 512) s_red[t] = g_w[t] * s_xw[t];
    __syncthreads();
    for (int s = 256; s > 0; s >>= 1) {
        if (t < s) s_red[t] += s_red[t + s];
        __syncthreads();
    }
    float g = s_red[0] + g_b[0];
    g = (g > 0.0f) ? g : 0.0f;                     // relu
    if (t < 512) s_x[t] = (1.0f - g) * s_xw[t] + g * s_xc[t];
    __syncthreads();

    for (int l = 0; l < 2; ++l) {
        const float* inp = (l == 0) ? s_x : s_h[0];
        const float* Wi = Wih + (size_t)l * 2048 * 512;
        const float* Wh = Whh + (size_t)l * 2048 * 512;
#pragma unroll
        for (int pp = 0; pp < 2; ++pp) {
            const int n0a = (wave + pp * 64) * 16;
            const int n0b = (wave + pp * 64 + 32) * 16;
            float ga, gb;
            wave_gates2(Wi, Wh, 512, n0a, n0b, inp, s_h[l], 512, &ga, &gb);
            if (lane < 16) {
                s_gates[n0a + lane] = ga;
                s_gates[n0b + lane] = gb;
            }
        }
        __syncthreads();
        if (t < 512) {
            const int b0 = l * 2048;
            float gi = s_gates[t]        + bih[b0 + t]        + bhh[b0 + t];
            float gf = s_gates[512 + t]  + bih[b0 + 512 + t]  + bhh[b0 + 512 + t];
            float gg = s_gates[1024 + t] + bih[b0 + 1024 + t] + bhh[b0 + 1024 + t];
            float go = s_gates[1536 + t] + bih[b0 + 1536 + t] + bhh[b0 + 1536 + t];
            float i = sigmoidf_(gi);
            float f = sigmoidf_(gf);
            float gv = tanhf(gg);
            float o = sigmoidf_(go);
            float c = f * s_c[l][t] + i * gv;
            s_c[l][t] = c;
            s_h[l][t] = o * tanhf(c);
        }
        __syncthreads();
    }

    if (t < 512) {
        out[OFF_WH + t]       = s_h[0][t];
        out[OFF_WH + 512 + t] = s_h[1][t];
        out[OFF_WC + t]       = s_c[0][t];
        out[OFF_WC + 512 + t] = s_c[1][t];
        y_ws[t] = s_h[1][t];
    }
}

// ---------------------------------------------------------------------------
// Kernel 3: decoder GEMV  y[512] x dec_W[50000 x 512]^T + dec_b.
// 102 MB streamed once -> non-temporal loads; 3125 row-tiles of 16,
// 391 blocks x 8 waves.  HBM-bound: ~4.4 us at 23.3 TB/s.
// ---------------------------------------------------------------------------
__global__ __launch_bounds__(256) void decoder_kernel(
    const float* __restrict__ dec_W, const float* __restrict__ dec_b,
    const float* __restrict__ y, float* __restrict__ out) {
    __shared__ float sy[512];
    const int t = (int)threadIdx.x;
    sy[t] = y[t];
    sy[256 + t] = y[256 + t];
    __syncthreads();

    const int wave = t >> 5;
    const int lane = t & 31;
    const int chunk = (int)blockIdx.x * 8 + wave;   // wave-uniform
    if (chunk < 3125) {                             // 3125*16 == 50000 exactly
        const int n0 = chunk * 16;
        float r = wmma_mv_split2<true>(dec_W, 512, n0, sy, 512);
        if (lane < 16) out[n0 + lane] = r + dec_b[n0 + lane];
    }
}

// ---------------------------------------------------------------------------
extern "C" void kernel_launch(void* const* d_in, const int* in_sizes, int n_in,
                              void* d_out, int out_size, void* d_ws, size_t ws_size,
                              hipStream_t stream) {
    (void)in_sizes; (void)n_in; (void)out_size; (void)ws_size;

    const int*   x_word    = (const int*)d_in[0];
    const int*   x_char    = (const int*)d_in[1];
    const float* h_word_h  = (const float*)d_in[2];
    const float* h_word_c  = (const float*)d_in[3];
    const float* h_char_h  = (const float*)d_in[4];
    const float* h_char_c  = (const float*)d_in[5];
    const float* word_emb  = (const float*)d_in[6];
    const float* word_Wih  = (const float*)d_in[7];
    const float* word_Whh  = (const float*)d_in[8];
    const float* word_bih  = (const float*)d_in[9];
    const float* word_bhh  = (const float*)d_in[10];
    const float* dec_W     = (const float*)d_in[11];
    const float* dec_b     = (const float*)d_in[12];
    const float* char_emb  = (const float*)d_in[13];
    const float* char_Wih  = (const float*)d_in[14];
    const float* char_Whh  = (const float*)d_in[15];
    const float* char_bih  = (const float*)d_in[16];
    const float* char_bhh  = (const float*)d_in[17];
    const float* c2e       = (const float*)d_in[18];
    const float* g_w       = (const float*)d_in[19];
    const float* g_b       = (const float*)d_in[20];

    float* out  = (float*)d_out;
    float* y_ws = (float*)d_ws;   // 512 floats of scratch for top hidden state

    char_lstm_kernel<<<1, 1024, 0, stream>>>(x_char, h_char_h, h_char_c,
                                             char_emb, char_Wih, char_Whh,
                                             char_bih, char_bhh, out);

    word_lstm_kernel<<<1, 1024, 0, stream>>>(x_word, h_word_h, h_word_c,
                                             word_emb, word_Wih, word_Whh,
                                             word_bih, word_bhh,
                                             c2e, g_w, g_b, out, y_ws);

    decoder_kernel<<<391, 256, 0, stream>>>(dec_W, dec_b, y_ws, out);
}